// GraphAttentionLayer_90795608637866
// MI455X (gfx1250) — compile-verified
//
#include <hip/hip_runtime.h>
#include <hip/hip_bf16.h>
#include <math.h>

typedef __attribute__((ext_vector_type(2))) float v2f;
typedef __attribute__((ext_vector_type(8))) float v8f;

#define B_     8
#define N_     2048
#define FIN_   128
#define FOUT_  64
#define ALPHA_ 0.2f
#define NEG_BIG_ (-9.0e15f)

// ---------------------------------------------------------------------------
// Kernel 1: wh = x @ W  (per-row), s1 = wh . a1, s2 = wh . a2
// One 64-thread block (2 waves) per row; x row staged in LDS.
// ---------------------------------------------------------------------------
__global__ __launch_bounds__(64)
void gat_wh_kernel(const float* __restrict__ x, const float* __restrict__ w,
                   const float* __restrict__ w2, float* __restrict__ wh,
                   float* __restrict__ s1, float* __restrict__ s2) {
  __shared__ float xs[FIN_];
  __shared__ float r1[64];
  __shared__ float r2[64];
  const int row = blockIdx.x;      // = b*N + i
  const int t   = threadIdx.x;     // output feature o
  xs[t]      = x[(size_t)row * FIN_ + t];
  xs[t + 64] = x[(size_t)row * FIN_ + t + 64];
  __syncthreads();
  float acc = 0.f;
#pragma unroll
  for (int f = 0; f < FIN_; ++f) acc = fmaf(xs[f], w[f * FOUT_ + t], acc);
  wh[(size_t)row * FOUT_ + t] = acc;
  r1[t] = acc * w2[t];           // a1 = weight2[:64]
  r2[t] = acc * w2[FOUT_ + t];   // a2 = weight2[64:]
  __syncthreads();
  for (int off = 32; off > 0; off >>= 1) {
    if (t < off) { r1[t] += r1[t + off]; r2[t] += r2[t + off]; }
    __syncthreads();
  }
  if (t == 0) { s1[row] = r1[0]; s2[row] = r2[0]; }
}

// ---------------------------------------------------------------------------
// Kernel 2: per-row softmax stats.  rowmax[b,i], rowinv[b,i] = 1/sum(exp).
// One 256-thread block per row; e values cached in LDS (single adj read).
// ---------------------------------------------------------------------------
__global__ __launch_bounds__(256)
void gat_stats_kernel(const int* __restrict__ adj, const float* __restrict__ s1,
                      const float* __restrict__ s2, float* __restrict__ rmaxg,
                      float* __restrict__ rinvg) {
  __shared__ float ebuf[N_];
  __shared__ float red[256];
  const int row = blockIdx.x;            // = b*N + i
  const int b   = row >> 11;             // N_ == 2048
  const int t   = threadIdx.x;
  const float  s1v   = s1[row];
  const size_t abase = (size_t)row * N_;
  const int    s2b   = b * N_;
  float m = -INFINITY;
  for (int j = t; j < N_; j += 256) {
    int   a = adj[abase + j];
    float e = s1v + s2[s2b + j];
    e = (e > 0.f) ? e : (ALPHA_ * e);    // leaky relu
    e = (a > 0) ? e : NEG_BIG_;          // adjacency mask
    ebuf[j] = e;
    m = fmaxf(m, e);
  }
  red[t] = m;
  __syncthreads();
  for (int off = 128; off > 0; off >>= 1) {
    if (t < off) red[t] = fmaxf(red[t], red[t + off]);
    __syncthreads();
  }
  const float rmax = red[0];
  __syncthreads();
  float s = 0.f;
  for (int j = t; j < N_; j += 256) s += __expf(ebuf[j] - rmax);
  red[t] = s;
  __syncthreads();
  for (int off = 128; off > 0; off >>= 1) {
    if (t < off) red[t] += red[t + off];
    __syncthreads();
  }
  if (t == 0) { rmaxg[row] = rmax; rinvg[row] = 1.f / red[0]; }
}

// ---------------------------------------------------------------------------
// Kernel 3: out = elu(softmax(e) @ wh) via V_WMMA_F32_16X16X4_F32.
// Block = 128 threads (4 waves) owns a 16-row i-tile x 64 outputs.
// Wave w owns output N-columns [16w, 16w+16).  Per 128-wide j-chunk:
//   - coalesced adj loads -> P tile (16x132, padded) in LDS
//   - coalesced b128 loads stage wh chunk (128x72, padded) in LDS
//   - 32 accumulating K=4 fp32 WMMAs per wave, A and B both from LDS
// ---------------------------------------------------------------------------
__global__ __launch_bounds__(128)
void gat_attn_mm_kernel(const int* __restrict__ adj, const float* __restrict__ wh,
                        const float* __restrict__ s1, const float* __restrict__ s2,
                        const float* __restrict__ rmaxg, const float* __restrict__ rinvg,
                        float* __restrict__ out) {
  constexpr int PSTR = 132;   // P row stride: rows land 4 banks apart -> conflict-free
  constexpr int WSTR = 72;    // wh row stride: K-half rows (d=2) 16 banks apart -> conflict-free
  __shared__ __align__(16) float plds[16 * PSTR];   //  8448 B
  __shared__ __align__(16) float wlds[128 * WSTR];  // 36864 B
  __shared__ float sm_s1[16], sm_max[16], sm_inv[16];
  const int t    = threadIdx.x;
  const int b    = blockIdx.x >> 7;            // 128 i-tiles per batch
  const int i0   = (blockIdx.x & 127) << 4;
  const int wave = t >> 5;
  const int lane = t & 31;
  const int mrow = lane & 15;                  // A row / B,D column within tile
  const int kh   = lane >> 4;                  // K half-select (per ISA 16x4 layout)
  const int n0   = wave << 4;

  if (t < 16) {
    int gr = b * N_ + i0 + t;
    sm_s1[t]  = s1[gr];
    sm_max[t] = rmaxg[gr];
    sm_inv[t] = rinvg[gr];
  }
  __syncthreads();

  v8f acc = {};
  const size_t adj_r0  = ((size_t)b * N_ + i0) * N_;
  const size_t wh_base = (size_t)b * N_ * FOUT_;
  const int    s2b     = b * N_;
  const int    sr0     = t >> 4;               // staging: row sub-index 0..7
  const int    sc4     = (t & 15) << 2;        // staging: column group 0,4,..,60

  for (int j0 = 0; j0 < N_; j0 += 128) {
    // ---- stage wh chunk (128 x 64) into LDS, b128-coalesced ----
    const float* whsrc = wh + wh_base + (size_t)j0 * FOUT_;
#pragma unroll
    for (int it = 0; it < 16; ++it) {
      const int r = it * 8 + sr0;
      *(float4*)&wlds[r * WSTR + sc4] = *(const float4*)&whsrc[r * FOUT_ + sc4];
    }
    // ---- build P tile (16 x 128) in LDS, fully coalesced adj reads ----
    const float s2v = s2[s2b + j0 + t];
#pragma unroll
    for (int r = 0; r < 16; ++r) {
      int   a = adj[adj_r0 + (size_t)r * N_ + j0 + t];
      float e = sm_s1[r] + s2v;
      e = (e > 0.f) ? e : (ALPHA_ * e);
      e = (a > 0) ? e : NEG_BIG_;
      plds[r * PSTR + t] = __expf(e - sm_max[r]) * sm_inv[r];
    }
    // prefetch next adj chunk into caches (global_prefetch_b8)
    if (j0 + 128 < N_) {
#pragma unroll
      for (int r = 0; r < 16; ++r)
        __builtin_prefetch(&adj[adj_r0 + (size_t)r * N_ + j0 + 128 + t], 0, 0);
    }
    __syncthreads();

    // ---- 32 WMMA steps of K=4 over this chunk, operands from LDS ----
#pragma unroll 8
    for (int jj = 0; jj < 128; jj += 4) {
      // A (16x4 f32): vgpr0 = K=jj+2*kh, vgpr1 = K=jj+1+2*kh, row = mrow
      v2f A = *(const v2f*)&plds[mrow * PSTR + jj + 2 * kh];
      // B (4x16 f32): vgpr0 = K row jj+2*kh, vgpr1 = K row jj+1+2*kh, col = mrow
      v2f Bv;
      Bv.x = wlds[(jj + 2 * kh) * WSTR + n0 + mrow];
      Bv.y = wlds[(jj + 2 * kh + 1) * WSTR + n0 + mrow];
      acc = __builtin_amdgcn_wmma_f32_16x16x4_f32(false, A, false, Bv,
                                                  (short)0, acc, false, false);
    }
    __syncthreads();
  }

  // ---- epilogue: ELU + store (C/D layout: vgpr k -> M = k + 8*kh, N = mrow) ----
#pragma unroll
  for (int k = 0; k < 8; ++k) {
    float v = acc[k];
    v = (v > 0.f) ? v : expm1f(v);
    out[((size_t)(b * N_ + i0 + k + 8 * kh)) * FOUT_ + n0 + mrow] = v;
  }
}

// ---------------------------------------------------------------------------
extern "C" void kernel_launch(void* const* d_in, const int* in_sizes, int n_in,
                              void* d_out, int out_size, void* d_ws, size_t ws_size,
                              hipStream_t stream) {
  const float* x   = (const float*)d_in[0];
  const int*   adj = (const int*)d_in[1];
  const float* w   = (const float*)d_in[2];
  const float* w2  = (const float*)d_in[3];
  float* out = (float*)d_out;

  float* ws   = (float*)d_ws;
  float* wh   = ws;                                   // B*N*FOUT   = 1,048,576 f
  float* s1   = wh + (size_t)B_ * N_ * FOUT_;         // B*N        =    16,384 f
  float* s2   = s1 + B_ * N_;
  float* rmax = s2 + B_ * N_;
  float* rinv = rmax + B_ * N_;                       // total ~4.25 MB

  gat_wh_kernel<<<B_ * N_, 64, 0, stream>>>(x, w, w2, wh, s1, s2);
  gat_stats_kernel<<<B_ * N_, 256, 0, stream>>>(adj, s1, s2, rmax, rinv);
  gat_attn_mm_kernel<<<(B_ * N_) / 16, 128, 0, stream>>>(adj, wh, s1, s2, rmax, rinv, out);
}